// IGNN_solver_24919400251504
// MI455X (gfx1250) — compile-verified
//
#include <hip/hip_runtime.h>
#include <hip/hip_bf16.h>
#include <math.h>

typedef float v2f __attribute__((ext_vector_type(2)));
typedef float v8f __attribute__((ext_vector_type(8)));

#define HID 128
#define KAPPA 0.9f
#define POW_ITERS 50
#define AND_M 5
#define AND_LAM 1e-4f
#define FP_THRESHOLD 20
#define NB_GRAM 256

// ---------------- utility fills ----------------
__global__ void fill_f32(float* p, float v, long n) {
    long i = blockIdx.x * (long)blockDim.x + threadIdx.x;
    long st = (long)gridDim.x * blockDim.x;
    for (; i < n; i += st) p[i] = v;
}
__global__ void fill_u32(unsigned* p, unsigned v, long n) {
    long i = blockIdx.x * (long)blockDim.x + threadIdx.x;
    long st = (long)gridDim.x * blockDim.x;
    for (; i < n; i += st) p[i] = v;
}

// ---------------- CSR build ----------------
__global__ void edge_count(const int* __restrict__ rows, unsigned* __restrict__ cnt, int e) {
    int i = blockIdx.x * blockDim.x + threadIdx.x;
    if (i < e) atomicAdd(&cnt[rows[i]], 1u);
}

__global__ void scan_counts(const unsigned* __restrict__ cnts, unsigned* __restrict__ ptro,
                            unsigned* __restrict__ fill, int n) {
    __shared__ unsigned s[1024];
    __shared__ unsigned carry;
    if (threadIdx.x == 0) carry = 0u;
    __syncthreads();
    for (int base = 0; base < n; base += 1024) {
        int i = base + (int)threadIdx.x;
        unsigned x = (i < n) ? cnts[i] : 0u;
        s[threadIdx.x] = x;
        __syncthreads();
        for (int off = 1; off < 1024; off <<= 1) {
            unsigned t = (threadIdx.x >= (unsigned)off) ? s[threadIdx.x - off] : 0u;
            __syncthreads();
            s[threadIdx.x] += t;
            __syncthreads();
        }
        unsigned excl = s[threadIdx.x] - x + carry;
        if (i < n) { ptro[i] = excl; fill[i] = excl; }
        __syncthreads();
        if (threadIdx.x == 1023) carry += s[1023];
        __syncthreads();
    }
    if (threadIdx.x == 0) ptro[n] = carry;
}

__global__ void edge_scatter(const int* __restrict__ rows, const int* __restrict__ cols,
                             const float* __restrict__ w, unsigned* __restrict__ fill,
                             unsigned* __restrict__ ocol, float* __restrict__ ow, int e) {
    int i = blockIdx.x * blockDim.x + threadIdx.x;
    if (i < e) {
        unsigned p = atomicAdd(&fill[rows[i]], 1u);
        ocol[p] = (unsigned)cols[i];
        ow[p]   = w[i];
    }
}

// ---------------- power iteration ----------------
__global__ void pow_gather(const unsigned* __restrict__ ptr, const unsigned* __restrict__ cols,
                           const float* __restrict__ w, const float* __restrict__ v,
                           float* __restrict__ u, float* __restrict__ part, int n) {
    int r = blockIdx.x * blockDim.x + threadIdx.x;
    float s = 0.f;
    if (r < n) {
        unsigned e0 = ptr[r], e1 = ptr[r + 1];
        for (unsigned e = e0; e < e1; ++e) s += fabsf(w[e]) * v[cols[e]];
        u[r] = s;
    }
    __shared__ float red[256];
    red[threadIdx.x] = s * s;
    __syncthreads();
    for (int st = 128; st > 0; st >>= 1) {
        if ((int)threadIdx.x < st) red[threadIdx.x] += red[threadIdx.x + st];
        __syncthreads();
    }
    if (threadIdx.x == 0) part[blockIdx.x] = red[0];
}

__global__ void pow_finish(const float* __restrict__ part, int nb, float* __restrict__ scal,
                           int storeRho) {
    __shared__ float red[256];
    float s = 0.f;
    for (int i = threadIdx.x; i < nb; i += 256) s += part[i];
    red[threadIdx.x] = s;
    __syncthreads();
    for (int st = 128; st > 0; st >>= 1) {
        if ((int)threadIdx.x < st) red[threadIdx.x] += red[threadIdx.x + st];
        __syncthreads();
    }
    if (threadIdx.x == 0) {
        float nrm = sqrtf(red[0]);
        scal[0] = nrm;
        if (storeRho) scal[1] = nrm + 1e-5f;
    }
}

__global__ void pow_scale(const float* __restrict__ u, const float* __restrict__ scal,
                          float* __restrict__ v, int n) {
    int i = blockIdx.x * blockDim.x + threadIdx.x;
    if (i < n) v[i] = u[i] / (scal[0] + 1e-12f);
}

// ---------------- L-inf (row L1-ball) projection, one block per row ----------------
__global__ void proj_norm_inf(const float* __restrict__ W, const float* __restrict__ scal,
                              float* __restrict__ Wp) {
    __shared__ float s[HID];
    __shared__ float cum[HID];
    __shared__ int ired[HID];
    const int row = blockIdx.x, t = threadIdx.x;
    const float vlim = KAPPA / scal[1];
    float w = W[row * HID + t];
    float a = fabsf(w);
    s[t] = a;
    __syncthreads();
    // ascending bitonic sort of 128 values with 128 threads
    for (int k = 2; k <= HID; k <<= 1) {
        for (int j = k >> 1; j > 0; j >>= 1) {
            int ixj = t ^ j;
            if (ixj > t) {
                float x = s[t], y = s[ixj];
                bool up = ((t & k) == 0);
                if (up ? (x > y) : (x < y)) { s[t] = y; s[ixj] = x; }
            }
            __syncthreads();
        }
    }
    float ud = s[HID - 1 - t];           // descending order
    cum[t] = ud;
    __syncthreads();
    // inclusive scan
    for (int off = 1; off < HID; off <<= 1) {
        float tm = (t >= off) ? cum[t - off] : 0.f;
        __syncthreads();
        cum[t] += tm;
        __syncthreads();
    }
    float css = cum[t] - vlim;
    ired[t] = (ud > css / (float)(t + 1)) ? 1 : 0;
    __syncthreads();
    for (int st = HID / 2; st > 0; st >>= 1) {
        if (t < st) ired[t] += ired[t + st];
        __syncthreads();
    }
    int rho_i = ired[0] - 1;
    float theta = (cum[rho_i] - vlim) / (float)(rho_i + 1);
    float total = cum[HID - 1];
    float thr = fmaxf(theta, 0.f);
    float sgn = (w > 0.f) ? 1.f : ((w < 0.f) ? -1.f : 0.f);
    float wproj = sgn * fmaxf(a - thr, 0.f);
    Wp[row * HID + t] = (total > vlim) ? wproj : w;
}

// ---------------- WMMA f32 GEMM: C[M,N] = A[M,K] * B (opt. B = VW^T) ----------------
// Compile-time NTILES/TRANSB/GUARD -> static acc regs. Software-pipelined fragment
// loads (double buffer). All addressing is base-pointer + 32-bit element offsets so
// loads lower to global_load (saddr+voffset), not flat_load (no DScnt coupling,
// no 64-bit per-lane address chains).
template <int NTILES, bool TRANSB, bool GUARD>
__global__ void wmma_gemm_f32(const float* __restrict__ A, const float* __restrict__ B,
                              float* __restrict__ C, int M, int N, int K) {
    const int lane  = threadIdx.x;                       // 0..31 (wave32)
    const int mtile = blockIdx.x * blockDim.y + threadIdx.y;
    if (mtile * 16 >= M) return;
    const int mbase  = mtile * 16;
    const int nbase0 = blockIdx.y * NTILES * 16;
    const int nlane  = lane & 15;
    const int half   = lane >> 4;                        // 0 or 1
    const int mrow   = mbase + nlane;                    // A: M = lane%16
    const int khalf  = half * 2;                         // A/B: K-half by lane>=16

    int  ncol[NTILES];
    bool nok[NTILES];
    int  bofs[NTILES];                                   // 32-bit element offsets into B
    #pragma unroll
    for (int t = 0; t < NTILES; ++t) {
        ncol[t] = nbase0 + t * 16 + nlane;
        nok[t]  = GUARD ? (ncol[t] < N) : true;
        int nc  = nok[t] ? ncol[t] : 0;                  // safe offset for dead tiles
        bofs[t] = TRANSB ? (nc * K + khalf) : (khalf * N + nc);
    }
    const int bstep = TRANSB ? 4 : 4 * N;
    const int aofs0 = mrow * K + khalf;                  // 32-bit element offset into A

    v8f acc[NTILES] = {};

    // prologue: fragments for k = 0
    v2f a_c = *(const v2f*)(A + aofs0);
    v2f b_c[NTILES];
    #pragma unroll
    for (int t = 0; t < NTILES; ++t) {
        if (!GUARD || nok[t]) {
            if (TRANSB) {
                b_c[t] = *(const v2f*)(B + bofs[t]);
            } else {
                b_c[t].x = B[bofs[t]];
                b_c[t].y = B[bofs[t] + N];
            }
        } else {
            b_c[t].x = 0.f; b_c[t].y = 0.f;
        }
        bofs[t] += bstep;
    }

    // steady state: issue k+4 loads, then k's WMMAs (waits cover only older loads)
    for (int k = 4; k < K; k += 4) {
        v2f a_n = *(const v2f*)(A + aofs0 + k);
        v2f b_n[NTILES];
        #pragma unroll
        for (int t = 0; t < NTILES; ++t) {
            if (!GUARD || nok[t]) {
                if (TRANSB) {
                    b_n[t] = *(const v2f*)(B + bofs[t]);
                } else {
                    b_n[t].x = B[bofs[t]];
                    b_n[t].y = B[bofs[t] + N];
                }
            } else {
                b_n[t].x = 0.f; b_n[t].y = 0.f;
            }
            bofs[t] += bstep;
        }
        #pragma unroll
        for (int t = 0; t < NTILES; ++t)
            acc[t] = __builtin_amdgcn_wmma_f32_16x16x4_f32(
                false, a_c, false, b_c[t], (short)0, acc[t], false, false);
        a_c = a_n;
        #pragma unroll
        for (int t = 0; t < NTILES; ++t) b_c[t] = b_n[t];
    }
    // epilogue: last K-step
    #pragma unroll
    for (int t = 0; t < NTILES; ++t)
        acc[t] = __builtin_amdgcn_wmma_f32_16x16x4_f32(
            false, a_c, false, b_c[t], (short)0, acc[t], false, false);

    const int mo = half * 8;                             // D: M = vgpr + 8*(lane>=16)
    #pragma unroll
    for (int t = 0; t < NTILES; ++t) {
        if (GUARD && !nok[t]) continue;
        const int cbase = (mbase + mo) * N + ncol[t];
        #pragma unroll
        for (int j = 0; j < 8; ++j)
            C[cbase + j * N] = acc[t][j];
    }
}

// ---------------- gather spmm over 128 columns: out[r] = (addIn[r]) + sum w*Y[col] ----------------
__global__ void spmm_rows(const unsigned* __restrict__ ptr, const unsigned* __restrict__ cols,
                          const float* __restrict__ wv, const float* __restrict__ Yin,
                          const float* __restrict__ addIn, float* __restrict__ outp,
                          int doRelu) {
    __shared__ unsigned scol[HID];
    __shared__ float    swt[HID];
    const int r = blockIdx.x;
    const int c = threadIdx.x;
    const unsigned s0 = ptr[r], s1 = ptr[r + 1];
    float acc = addIn ? addIn[(size_t)r * HID + c] : 0.f;
    for (unsigned base = s0; base < s1; base += HID) {
        unsigned idx = base + (unsigned)c;
        if (idx < s1) { scol[c] = cols[idx]; swt[c] = wv[idx]; }
        __syncthreads();
        unsigned cnt = s1 - base;
        if (cnt > (unsigned)HID) cnt = HID;
        for (unsigned i = 0; i < cnt; ++i)
            acc += swt[i] * Yin[(size_t)scol[i] * HID + c];
        __syncthreads();
    }
    if (doRelu) acc = fmaxf(acc, 0.f);
    outp[(size_t)r * HID + c] = acc;
}

__global__ void relu_copy(const float* __restrict__ in, float* __restrict__ outp, long d) {
    long i = blockIdx.x * (long)blockDim.x + threadIdx.x;
    long st = (long)gridDim.x * blockDim.x;
    for (; i < d; i += st) outp[i] = fmaxf(in[i], 0.f);
}

// ---------------- Anderson: Gram of G = F - X (pairwise dots), fixed grid ----------------
__global__ void gram_partial(const float* __restrict__ Xh, const float* __restrict__ Fh,
                             long d, int nk, float* __restrict__ part) {
    float acc[15];
    #pragma unroll
    for (int p = 0; p < 15; ++p) acc[p] = 0.f;
    long i = blockIdx.x * (long)blockDim.x + threadIdx.x;
    long st = (long)gridDim.x * blockDim.x;
    for (; i < d; i += st) {
        float g[5];
        #pragma unroll
        for (int q = 0; q < 5; ++q)
            g[q] = (q < nk) ? (Fh[(size_t)q * d + i] - Xh[(size_t)q * d + i]) : 0.f;
        int p = 0;
        #pragma unroll
        for (int a = 0; a < 5; ++a)
            #pragma unroll
            for (int b = a; b < 5; ++b) { acc[p] += g[a] * g[b]; ++p; }
    }
    __shared__ float red[256];
    for (int p = 0; p < 15; ++p) {
        red[threadIdx.x] = acc[p];
        __syncthreads();
        for (int stp = 128; stp > 0; stp >>= 1) {
            if ((int)threadIdx.x < stp) red[threadIdx.x] += red[threadIdx.x + stp];
            __syncthreads();
        }
        if (threadIdx.x == 0) part[p * gridDim.x + blockIdx.x] = red[0];
        __syncthreads();
    }
}

// reduce partials, build bordered system H, Gauss-Jordan solve -> alpha[0..nk)
__global__ void gram_solve(const float* __restrict__ part, int nb, int nk, float lam,
                           float* __restrict__ alpha) {
    if (threadIdx.x != 0 || blockIdx.x != 0) return;
    float G[5][5];
    int p = 0;
    for (int i = 0; i < 5; ++i)
        for (int j = i; j < 5; ++j) {
            float s = 0.f;
            if (i < nk && j < nk)
                for (int b = 0; b < nb; ++b) s += part[p * nb + b];
            G[i][j] = s; G[j][i] = s;
            ++p;
        }
    const int m = nk + 1;
    float H[6][7];
    for (int r = 0; r < m; ++r)
        for (int c = 0; c <= m; ++c) H[r][c] = 0.f;
    for (int i = 1; i < m; ++i) { H[0][i] = 1.f; H[i][0] = 1.f; }
    for (int i = 0; i < nk; ++i)
        for (int j = 0; j < nk; ++j)
            H[i + 1][j + 1] = G[i][j] + ((i == j) ? lam : 0.f);
    H[0][m] = 1.f;  // rhs = e0
    for (int c = 0; c < m; ++c) {
        int piv = c; float best = fabsf(H[c][c]);
        for (int r = c + 1; r < m; ++r) {
            float v = fabsf(H[r][c]);
            if (v > best) { best = v; piv = r; }
        }
        if (piv != c)
            for (int q = 0; q <= m; ++q) { float t = H[c][q]; H[c][q] = H[piv][q]; H[piv][q] = t; }
        float inv = 1.f / H[c][c];
        for (int r = 0; r < m; ++r) {
            if (r == c) continue;
            float f = H[r][c] * inv;
            for (int q = c; q <= m; ++q) H[r][q] -= f * H[c][q];
        }
    }
    for (int i = 0; i < nk; ++i) alpha[i] = H[i + 1][m] / H[i + 1][i + 1];
}

__global__ void combine(const float* __restrict__ Fh, const float* __restrict__ al,
                        long d, int nk, float* __restrict__ outp) {
    float a[5];
    #pragma unroll
    for (int i = 0; i < 5; ++i) a[i] = (i < nk) ? al[i] : 0.f;
    long i = blockIdx.x * (long)blockDim.x + threadIdx.x;
    long st = (long)gridDim.x * blockDim.x;
    for (; i < d; i += st) {
        float s = 0.f;
        #pragma unroll
        for (int j = 0; j < 5; ++j)
            if (j < nk) s += a[j] * Fh[(size_t)j * d + i];
        outp[i] = s;
    }
}

// ===================================================================================
extern "C" void kernel_launch(void* const* d_in, const int* in_sizes, int n_in,
                              void* d_out, int out_size, void* d_ws, size_t ws_size,
                              hipStream_t stream) {
    const float* U    = (const float*)d_in[0];
    const int*   eidx = (const int*)d_in[1];     // (2, e) int32: rows then cols
    const float* ew   = (const float*)d_in[2];
    const float* W    = (const float*)d_in[3];
    const float* Bm   = (const float*)d_in[4];   // (256, 128)
    const float* VW   = (const float*)d_in[5];   // (nclass, 128)
    float* out = (float*)d_out;

    const int n = in_sizes[0] / 256;
    const int e = in_sizes[2];
    const int nclass = in_sizes[5] / HID;
    const long d = (long)n * HID;

    char* ws = (char*)d_ws;
    size_t off = 0;
    auto alloc = [&](size_t bytes) -> char* {
        char* p = ws + off;
        off += (bytes + 255) & ~(size_t)255;
        return p;
    };
    unsigned* counts   = (unsigned*)alloc((size_t)n * 4);
    unsigned* csr_ptr  = (unsigned*)alloc((size_t)(n + 1) * 4);
    unsigned* csr_fill = (unsigned*)alloc((size_t)n * 4);
    unsigned* csr_col  = (unsigned*)alloc((size_t)e * 4);
    float*    csr_w    = (float*)alloc((size_t)e * 4);
    float*    vvec     = (float*)alloc((size_t)n * 4);
    float*    uvec     = (float*)alloc((size_t)n * 4);
    float*    powp     = (float*)alloc((size_t)((n + 255) / 256) * 4);
    float*    scal     = (float*)alloc(256);
    float*    Wp       = (float*)alloc((size_t)HID * HID * 4);
    float*    AUB      = (float*)alloc((size_t)d * 4);
    float*    Ybuf     = (float*)alloc((size_t)d * 4);
    float*    Xh       = (float*)alloc((size_t)5 * d * 4);
    float*    Fh       = (float*)alloc((size_t)5 * d * 4);
    float*    gpart    = (float*)alloc((size_t)15 * NB_GRAM * 4);
    float*    alpha    = (float*)alloc(256);
    (void)ws_size; (void)n_in; (void)out_size;

    const int TB = 256;
    const int nbE = (e + TB - 1) / TB;
    const int nbN = (n + TB - 1) / TB;
    const int mtiles = n / 16;                       // 30000/16 = 1875
    const dim3 gemmBlk(32, 4);
    const dim3 gemmGrd2((mtiles + 3) / 4, 2);        // N=128, 4 tiles/wave
    const dim3 gemmGrdC((mtiles + 3) / 4, 1);        // N=nclass(<=48), 3 tiles/wave

    // ---- CSR build ----
    fill_u32<<<nbN, TB, 0, stream>>>(counts, 0u, n);
    edge_count<<<nbE, TB, 0, stream>>>(eidx, counts, e);
    scan_counts<<<1, 1024, 0, stream>>>(counts, csr_ptr, csr_fill, n);
    edge_scatter<<<nbE, TB, 0, stream>>>(eidx, eidx + e, ew, csr_fill, csr_col, csr_w, e);

    // ---- spectral radius via power iteration on |A| ----
    fill_f32<<<nbN, TB, 0, stream>>>(vvec, 1.0f / sqrtf((float)n), n);
    for (int it = 0; it < POW_ITERS; ++it) {
        pow_gather<<<nbN, TB, 0, stream>>>(csr_ptr, csr_col, csr_w, vvec, uvec, powp, n);
        pow_finish<<<1, 256, 0, stream>>>(powp, nbN, scal, (it == POW_ITERS - 1) ? 1 : 0);
        pow_scale<<<nbN, TB, 0, stream>>>(uvec, scal, vvec, n);
    }

    // ---- project W rows onto L1 ball of radius kappa/rho ----
    proj_norm_inf<<<HID, HID, 0, stream>>>(W, scal, Wp);

    // ---- A_U_B = spmm(U @ B)  (spmm commutes with right-multiplication) ----
    wmma_gemm_f32<4, false, false><<<gemmGrd2, gemmBlk, 0, stream>>>(U, Bm, Ybuf, n, HID, 256);
    spmm_rows<<<n, HID, 0, stream>>>(csr_ptr, csr_col, csr_w, Ybuf, nullptr, AUB, 0);

    // ---- Anderson acceleration ----
    // X0 = 0 ; F0 = relu(AUB)
    fill_f32<<<4096, TB, 0, stream>>>(Xh, 0.f, d);
    relu_copy<<<4096, TB, 0, stream>>>(AUB, Fh, d);
    // X1 = F0 ; F1 = f(F0)
    hipMemcpyAsync(Xh + d, Fh, (size_t)d * 4, hipMemcpyDeviceToDevice, stream);
    wmma_gemm_f32<4, false, false><<<gemmGrd2, gemmBlk, 0, stream>>>(Fh, Wp, Ybuf, n, HID, HID);
    spmm_rows<<<n, HID, 0, stream>>>(csr_ptr, csr_col, csr_w, Ybuf, AUB, Fh + d, 1);

    for (int k = 2; k < FP_THRESHOLD; ++k) {
        int nk = (k < AND_M) ? k : AND_M;
        gram_partial<<<NB_GRAM, 256, 0, stream>>>(Xh, Fh, d, nk, gpart);
        gram_solve<<<1, 32, 0, stream>>>(gpart, NB_GRAM, nk, AND_LAM, alpha);
        float* Xs = Xh + (size_t)(k % AND_M) * d;
        float* Fs = Fh + (size_t)(k % AND_M) * d;
        combine<<<4096, TB, 0, stream>>>(Fh, alpha, d, nk, Xs);
        wmma_gemm_f32<4, false, false><<<gemmGrd2, gemmBlk, 0, stream>>>(Xs, Wp, Ybuf, n, HID, HID);
        spmm_rows<<<n, HID, 0, stream>>>(csr_ptr, csr_col, csr_w, Ybuf, AUB, Fs, 1);
    }

    // z_star = F[(threshold-1) % m] = F[4]
    float* z = Fh + (size_t)4 * d;
    // label = z @ VW^T  (transB), then append z_star
    wmma_gemm_f32<3, true, true><<<gemmGrdC, gemmBlk, 0, stream>>>(z, VW, out, n, nclass, HID);
    hipMemcpyAsync(out + (size_t)n * nclass, z, (size_t)d * 4,
                   hipMemcpyDeviceToDevice, stream);
}